// CTProjector3D_32306744000776
// MI455X (gfx1250) — compile-verified
//
#include <hip/hip_runtime.h>
#include <hip/hip_bf16.h>
#include <stdint.h>

// ---------------------------------------------------------------------------
// CT forward projector, MI455X (gfx1250).
// Memory-bound: 135MB t_sorted stream (TDM -> LDS, TH=NT so it does not evict
// the volume from L2), 64MB volume gathers (L2-resident: 192MB L2).
// Wave-per-ray for gather locality, wave32 shuffle reduction.
// CDNA5 paths: tensor_load_to_lds + s_wait_tensorcnt; NT hints on streams.
// No WMMA: there is no matmul structure, only a 3x3 matvec per segment.
// ---------------------------------------------------------------------------

#define CT_NX 256
#define CT_NY 256
#define CT_NZ 256
#define CT_NINT 129                 // 129 t-values -> 128 segments
#define CT_NSEG 128
#define RAYS_PER_BLOCK 64
#define THREADS_PER_BLOCK 256       // 8 waves (wave32)
#define RAYS_PER_WAVE 8

typedef __attribute__((ext_vector_type(4))) unsigned int v4u;
typedef __attribute__((ext_vector_type(4))) int          v4i;
typedef __attribute__((ext_vector_type(8))) int          v8i;

#if defined(__has_builtin)
#  if __has_builtin(__builtin_amdgcn_tensor_load_to_lds)
#    define CT_HAVE_TDM 1
#  endif
#endif

#if __has_include(<hip/amd_detail/amd_gfx1250_TDM.h>)
#  define CT_TDM_6ARG 1
#endif

// gfx1250 CPol: TH in [2:0]; load TH=1 -> NT (non-temporal), scope=WGP.
#define CT_CPOL_NT 1

__global__ __launch_bounds__(THREADS_PER_BLOCK)
void ct_fwd_kernel(const float* __restrict__ volume,
                   const float* __restrict__ t_sorted,
                   const float* __restrict__ M,
                   const float* __restrict__ bvec,
                   const float* __restrict__ src,
                   const float* __restrict__ dst,
                   float* __restrict__ out,
                   int n_ray)
{
    // 64 rays x 129 floats = 33,024 B of LDS (row stride 129 floats -> odd
    // DWORD stride -> bank-conflict-free across lanes).
    __shared__ float tbuf[RAYS_PER_BLOCK * CT_NINT];

    const int tid        = (int)threadIdx.x;
    const int block_ray0 = (int)blockIdx.x * RAYS_PER_BLOCK;
    const float* trow_g  = t_sorted + (size_t)block_ray0 * CT_NINT;

#if CT_HAVE_TDM
    // ---- TDM: DMA the 64x129 f32 tile of t_sorted into LDS (wave 0 only;
    //      TDM ignores EXEC, so gate whole-wave with a uniform branch). ----
    if (tid < 32) {
        unsigned long long ga = (unsigned long long)(uintptr_t)trow_g;
        unsigned int lds_addr = (unsigned int)(uintptr_t)(void*)&tbuf[0]; // low 32b of flat ptr = LDS offset

        // D# group 0: [1:0]=count=1, [63:32]=lds_addr, [120:64]=global addr,
        // [127:126]=type=2 ("image").
        v4u g0;
        g0[0] = 1u;
        g0[1] = lds_addr;
        g0[2] = (unsigned int)(ga & 0xFFFFFFFFull);
        g0[3] = (unsigned int)((ga >> 32) & 0x1FFFFFFull) | (2u << 30);

        // D# group 1: workgroup_mask=0 (not in a cluster), data_size=2 (4B),
        // tensor_dim0=129, tensor_dim1=64, tile_dim0=129, tile_dim1=64,
        // tile_dim2=0 (2D), tensor_dim0_stride=129, no padding/iteration.
        v8i g1;
        g1[0] = (int)(2u << 16);                 // data_size = 4 bytes
        g1[1] = (int)((unsigned)CT_NINT << 16);  // tensor_dim0[15:0]
        g1[2] = (int)((unsigned)RAYS_PER_BLOCK << 16); // tensor_dim0 hi=0 | tensor_dim1 lo
        g1[3] = (int)((unsigned)CT_NINT << 16);  // tensor_dim1 hi=0 | tile_dim0
        g1[4] = (int)RAYS_PER_BLOCK;             // tile_dim1 | tile_dim2=0
        g1[5] = (int)CT_NINT;                    // tensor_dim0_stride[31:0]
        g1[6] = 0;                               // stride hi | tensor_dim1_stride lo (unused, 2D)
        g1[7] = 0;

        v4i z4 = {0, 0, 0, 0};
#if defined(CT_TDM_6ARG)
        v8i z8 = {0, 0, 0, 0, 0, 0, 0, 0};
        __builtin_amdgcn_tensor_load_to_lds(g0, g1, z4, z4, z8, CT_CPOL_NT);
#else
        __builtin_amdgcn_tensor_load_to_lds(g0, g1, z4, z4, CT_CPOL_NT);
#endif
        __builtin_amdgcn_s_wait_tensorcnt(0);
    }
#else
    // Fallback: cooperative coalesced copy.
    for (int i = tid; i < RAYS_PER_BLOCK * CT_NINT; i += THREADS_PER_BLOCK)
        tbuf[i] = trow_g[i];
#endif
    __syncthreads();

    // ---- 3x3 inverse of M (uniform; a few dozen FLOPs, negligible). ----
    const float m00 = M[0], m01 = M[1], m02 = M[2];
    const float m10 = M[3], m11 = M[4], m12 = M[5];
    const float m20 = M[6], m21 = M[7], m22 = M[8];
    const float det = m00 * (m11 * m22 - m12 * m21)
                    - m01 * (m10 * m22 - m12 * m20)
                    + m02 * (m10 * m21 - m11 * m20);
    const float r   = 1.0f / det;
    const float i00 = (m11 * m22 - m12 * m21) * r;
    const float i01 = (m02 * m21 - m01 * m22) * r;
    const float i02 = (m01 * m12 - m02 * m11) * r;
    const float i10 = (m12 * m20 - m10 * m22) * r;
    const float i11 = (m00 * m22 - m02 * m20) * r;
    const float i12 = (m02 * m10 - m00 * m12) * r;
    const float i20 = (m10 * m21 - m11 * m20) * r;
    const float i21 = (m01 * m20 - m00 * m21) * r;
    const float i22 = (m00 * m11 - m01 * m10) * r;
    const float bx = bvec[0], by = bvec[1], bz = bvec[2];

    const int wave = tid >> 5;
    const int lane = tid & 31;

    // ---- wave-per-ray: lane l integrates segments l, l+32, l+64, l+96 ----
    for (int rr = 0; rr < RAYS_PER_WAVE; ++rr) {
        const int lray = wave * RAYS_PER_WAVE + rr;   // 0..63 within block
        const int ray  = block_ray0 + lray;
        if (ray >= n_ray) break;

        // src/dst are streamed exactly once -> NT so they don't evict the
        // volume working set from L2.
        const float sx = __builtin_nontemporal_load(src + ray * 3 + 0);
        const float sy = __builtin_nontemporal_load(src + ray * 3 + 1);
        const float sz = __builtin_nontemporal_load(src + ray * 3 + 2);
        const float dx = __builtin_nontemporal_load(dst + ray * 3 + 0) - sx;
        const float dy = __builtin_nontemporal_load(dst + ray * 3 + 1) - sy;
        const float dz = __builtin_nontemporal_load(dst + ray * 3 + 2) - sz;

        const float* trow = &tbuf[lray * CT_NINT];
        float acc = 0.0f;

#pragma unroll
        for (int k = 0; k < 4; ++k) {
            const int s = lane + 32 * k;              // 0..127
            const float t0 = trow[s];
            const float t1 = trow[s + 1];

            const float p0x = sx + t0 * dx, p0y = sy + t0 * dy, p0z = sz + t0 * dz;
            const float p1x = sx + t1 * dx, p1y = sy + t1 * dy, p1z = sz + t1 * dz;

            const bool infm = isinf(p0x) || isinf(p0y) || isinf(p0z) ||
                              isinf(p1x) || isinf(p1y) || isinf(p1z);

            const float ex = p1x - p0x, ey = p1y - p0y, ez = p1z - p0z;
            const float seg = sqrtf(ex * ex + ey * ey + ez * ez);

            const float mx = 0.5f * (p0x + p1x) - bx;
            const float my = 0.5f * (p0y + p1y) - by;
            const float mz = 0.5f * (p0z + p1z) - bz;

            // jnp.round == round-half-to-even -> v_rndne_f32
            const int ii = (int)rintf(i00 * mx + i01 * my + i02 * mz);
            const int jj = (int)rintf(i10 * mx + i11 * my + i12 * mz);
            const int kk = (int)rintf(i20 * mx + i21 * my + i22 * mz);

            const bool oob = ((unsigned)ii >= (unsigned)CT_NX) ||
                             ((unsigned)jj >= (unsigned)CT_NY) ||
                             ((unsigned)kk >= (unsigned)CT_NZ);
            const bool valid = !infm && !oob;

            const int flat = valid ? (ii * (CT_NY * CT_NZ) + jj * CT_NZ + kk) : 0;
            const float v = volume[flat];             // L2-resident gather (RT hint)
            acc += valid ? v * seg : 0.0f;
        }

        // wave32 reduction
        for (int off = 16; off > 0; off >>= 1)
            acc += __shfl_xor(acc, off, 32);
        if (lane == 0)
            __builtin_nontemporal_store(acc, out + ray);  // write-once stream
    }
}

extern "C" void kernel_launch(void* const* d_in, const int* in_sizes, int n_in,
                              void* d_out, int out_size, void* d_ws, size_t ws_size,
                              hipStream_t stream) {
    (void)n_in; (void)d_ws; (void)ws_size; (void)in_sizes;
    const float* volume   = (const float*)d_in[0];
    const float* t_sorted = (const float*)d_in[1];
    const float* M        = (const float*)d_in[2];
    const float* bvec     = (const float*)d_in[3];
    const float* src      = (const float*)d_in[4];
    const float* dst      = (const float*)d_in[5];
    float* out            = (float*)d_out;

    const int n_ray  = out_size;                                  // 262144
    const int blocks = (n_ray + RAYS_PER_BLOCK - 1) / RAYS_PER_BLOCK;

    hipLaunchKernelGGL(ct_fwd_kernel, dim3(blocks), dim3(THREADS_PER_BLOCK), 0, stream,
                       volume, t_sorted, M, bvec, src, dst, out, n_ray);
}